// EMAKMeansVectorQuantizer_58686433133154
// MI455X (gfx1250) — compile-verified
//
#include <hip/hip_runtime.h>
#include <hip/hip_bf16.h>
#include <math.h>

// ---------------------------------------------------------------------------
// EMA K-means Vector Quantizer forward, CDNA5 (gfx1250, wave32, WMMA fp32,
// async global->LDS double-buffered staging).
//   x      : [16, 512, 4096] f32
//   proj_w : [64, 512]       f32
//   proj_b : [64]            f32
//   embed  : [2048, 64]      f32
// Outputs (concat, f32): z_q_out [16,64,4096], loss [1], kldiv_r [16], ppl [1]
// ---------------------------------------------------------------------------

#define B_SZ    16
#define T_SZ    4096
#define C_SZ    512
#define E_SZ    64
#define K_SZ    2048
#define N_TOK   (B_SZ * T_SZ)          // 65536 tokens
#define COMMIT  0.25f

typedef float v2f __attribute__((ext_vector_type(2)));
typedef float v8f __attribute__((ext_vector_type(8)));

// workspace layout (in floats)
#define WS_H      0                        // h: N_TOK*64
#define WS_ENORM  (N_TOK * E_SZ)           // enorm: K_SZ
#define WS_QIDX   (WS_ENORM + K_SZ)        // q_idx: N_TOK ints
#define WS_HIST   (WS_QIDX + N_TOK)        // hist: K_SZ ints
#define WS_PART   (WS_HIST + K_SZ)         // partials: 1024 floats

// output layout (in floats)
#define OUT_Z     0
#define OUT_LOSS  (N_TOK * E_SZ)
#define OUT_KL    (OUT_LOSS + 1)
#define OUT_PPL   (OUT_KL + B_SZ)

// ------------------------------------------------------- async LDS helpers
// Per-lane global->LDS async copy of 8 bytes (a contiguous (c, c+1) K-pair).
// VDST operand carries the wave-relative LDS byte offset (low 32 bits of the
// flat pointer to a __shared__ object); tracked by ASYNCcnt.
__device__ __forceinline__ void async_b64_to_lds(unsigned lds_byte_off,
                                                 const float* gptr) {
    asm volatile("global_load_async_to_lds_b64 %0, %1, off"
                 :: "v"(lds_byte_off), "v"(gptr) : "memory");
}
__device__ __forceinline__ void wait_async0() {
    asm volatile("s_wait_asynccnt 0x0" ::: "memory");
}

// ---------------------------------------------------------------- zero hist
__global__ void vq_zero_hist(int* __restrict__ hist) {
    int i = blockIdx.x * blockDim.x + threadIdx.x;
    if (i < K_SZ) hist[i] = 0;
}

// ------------------------------------------------------- codeword sq-norms
__global__ void vq_enorm(const float* __restrict__ embed, float* __restrict__ enorm) {
    int k = blockIdx.x * blockDim.x + threadIdx.x;
    if (k >= K_SZ) return;
    const float* row = embed + k * E_SZ;
    float s = 0.f;
#pragma unroll
    for (int c = 0; c < E_SZ; ++c) s += row[c] * row[c];
    enorm[k] = s;
}

// ---------------------------------------------------------------------------
// Projection GEMM: h[n, e] = sum_c W[e,c] * x[b, c, t] + bias[e]
// 8 waves/block, 16-token M-tile per wave, K=512 in 8 chunks of 64.
// W chunk staged asynchronously into double-buffered paired LDS
// ( [cpair][e] pairs of (c, c+1) ) so B fragments are single ds_load_b64.
// ---------------------------------------------------------------------------
__device__ __forceinline__ void proj_stage(const float* __restrict__ W, int c0,
                                           float* buf, int tid) {
    const unsigned base = (unsigned)(size_t)buf;
#pragma unroll
    for (int i = 0; i < 8; ++i) {
        const int q  = i * 256 + tid;        // pair id 0..2047
        const int e  = q >> 5;               // 0..63
        const int cp = q & 31;               // c-pair 0..31
        async_b64_to_lds(base + (unsigned)((cp * E_SZ + e) * 8),
                         W + e * C_SZ + c0 + cp * 2);
    }
}

__global__ void __launch_bounds__(256)
vq_proj_wmma(const float* __restrict__ x, const float* __restrict__ W,
             const float* __restrict__ bias, float* __restrict__ h) {
    __shared__ float ldsW[2][64 * E_SZ];   // double-buffered 16 KB chunks

    const int tid  = threadIdx.x;
    const int wave = tid >> 5;
    const int lane = tid & 31;
    const int half = lane >> 4;        // 0: lanes 0-15, 1: lanes 16-31
    const int l16  = lane & 15;

    const int tok0 = blockIdx.x * 128 + wave * 16;   // flat token base
    const int b    = tok0 >> 12;                     // /4096
    const int t    = tok0 & 4095;
    const float* xb = x + (size_t)b * C_SZ * T_SZ + t;   // x[b, 0, t]

    v8f acc[4];
#pragma unroll
    for (int nt = 0; nt < 4; ++nt) acc[nt] = (v8f)0.f;

    proj_stage(W, 0, &ldsW[0][0], tid);
    wait_async0();
    __syncthreads();

    for (int cc = 0; cc < 8; ++cc) {          // K chunks of 64
        const int cur = cc & 1;
        if (cc + 1 < 8) proj_stage(W, (cc + 1) * 64, &ldsW[cur ^ 1][0], tid);

        const int c0 = cc * 64;
#pragma unroll
        for (int f = 0; f < 16; ++f) {            // 16 k-steps of 4
            // A fragment: lane m = l16, K in {half*2, half*2+1} of this step
            const int ca = c0 + f * 4 + half * 2;
            v2f a;
            a.x = xb[(size_t)(ca + 0) * T_SZ + l16];
            a.y = xb[(size_t)(ca + 1) * T_SZ + l16];
#pragma unroll
            for (int nt = 0; nt < 4; ++nt) {
                const int en = nt * 16 + l16;
                const v2f bfrag =
                    *(const v2f*)&ldsW[cur][((f * 2 + half) * E_SZ + en) * 2];
                acc[nt] = __builtin_amdgcn_wmma_f32_16x16x4_f32(
                    false, a, false, bfrag, (short)0, acc[nt], false, false);
            }
        }
        wait_async0();       // next chunk landed (overlapped with compute)
        __syncthreads();
    }

    // bias + store: D vgpr r -> row (r + half*8), col l16 within n-tile
#pragma unroll
    for (int nt = 0; nt < 4; ++nt) {
        const int e = nt * 16 + l16;
        const float bv = bias[e];
#pragma unroll
        for (int r = 0; r < 8; ++r) {
            const int tok = tok0 + r + half * 8;
            h[(size_t)tok * E_SZ + e] = acc[nt][r] + bv;
        }
    }
}

// ---------------------------------------------------------------------------
// Distance GEMM + fused argmin:  q_idx[n] = argmin_k (||e_k||^2 - 2 h_n.e_k)
// A fragments pre-scaled by -2; accumulator initialized with ||e_k||^2 splat,
// so each tile's result IS the (row-shifted) distance. Embed tiles staged
// async into double-buffered paired LDS (conflict-free b64 fragment reads).
// ---------------------------------------------------------------------------
__device__ __forceinline__ void dist_stage(const float* __restrict__ embed, int k0,
                                           float* buf, int tid) {
    const unsigned base = (unsigned)(size_t)buf;
#pragma unroll
    for (int i = 0; i < 2; ++i) {
        const int q  = i * 256 + tid;        // pair id 0..511
        const int cl = q >> 5;               // code 0..15
        const int cp = q & 31;               // c-pair 0..31
        async_b64_to_lds(base + (unsigned)((cp * 16 + cl) * 8),
                         embed + (size_t)(k0 + cl) * E_SZ + cp * 2);
    }
}

__global__ void __launch_bounds__(256)
vq_dist_argmin(const float* __restrict__ h, const float* __restrict__ embed,
               const float* __restrict__ enorm, int* __restrict__ qidx) {
    __shared__ float ldsE[2][64 * 16];     // double-buffered 4 KB tiles

    const int tid  = threadIdx.x;
    const int wave = tid >> 5;
    const int lane = tid & 31;
    const int half = lane >> 4;
    const int l16  = lane & 15;

    const int tok0 = blockIdx.x * 128 + wave * 16;
    const int row  = tok0 + l16;

    // preload A fragments for this wave's 16x64 h tile, scaled by -2
    v2f afrag[16];
#pragma unroll
    for (int f = 0; f < 16; ++f) {
        const int c = f * 4 + half * 2;
        v2f a = *(const v2f*)&h[(size_t)row * E_SZ + c];
        afrag[f].x = -2.f * a.x;
        afrag[f].y = -2.f * a.y;
    }

    float bestd[8];
    unsigned bestk[8];
#pragma unroll
    for (int r = 0; r < 8; ++r) { bestd[r] = __builtin_inff(); bestk[r] = 0u; }

    dist_stage(embed, 0, &ldsE[0][0], tid);
    wait_async0();
    __syncthreads();

    for (int j = 0; j < K_SZ / 16; ++j) {        // 128 code tiles
        const int cur = j & 1;
        if (j + 1 < K_SZ / 16) dist_stage(embed, (j + 1) * 16, &ldsE[cur ^ 1][0], tid);

        const int k0 = j * 16;
        const float en = enorm[k0 + l16];        // column bias, per lane
        v8f acc;
#pragma unroll
        for (int r = 0; r < 8; ++r) acc[r] = en; // C init = ||e||^2

#pragma unroll
        for (int f = 0; f < 16; ++f) {
            const v2f bfrag = *(const v2f*)&ldsE[cur][((f * 2 + half) * 16 + l16) * 2];
            acc = __builtin_amdgcn_wmma_f32_16x16x4_f32(
                false, afrag[f], false, bfrag, (short)0, acc, false, false);
        }

        const unsigned kc = (unsigned)(k0 + l16);
#pragma unroll
        for (int r = 0; r < 8; ++r) {
            if (acc[r] < bestd[r]) { bestd[r] = acc[r]; bestk[r] = kc; }
        }
        wait_async0();       // next tile landed (overlapped with compute)
        __syncthreads();
    }

    // per-row argmin across the 16 lanes of each half-wave (tie -> lower idx)
#pragma unroll
    for (int r = 0; r < 8; ++r) {
        float d = bestd[r];
        unsigned k = bestk[r];
#pragma unroll
        for (int off = 1; off < 16; off <<= 1) {
            float    od = __shfl_xor(d, off, 32);
            unsigned ok = __shfl_xor(k, off, 32);
            if (od < d || (od == d && ok < k)) { d = od; k = ok; }
        }
        if (l16 == 0) qidx[tok0 + r + half * 8] = (int)k;
    }
}

// ------------------------------------------------------------- histogram
__global__ void vq_hist(const int* __restrict__ qidx, int* __restrict__ hist) {
    int n = blockIdx.x * blockDim.x + threadIdx.x;
    if (n < N_TOK) atomicAdd(&hist[qidx[n]], 1);
}

// ---------------------------------------------------------------------------
// Gather z_q, transposed store to [B,E,T], per-block deterministic partial
// of the commitment SSE. One block per (b,e) row; coalesced T-direction;
// h and embed stay L2-resident (16.8 MB + 0.5 MB << 192 MB L2).
// ---------------------------------------------------------------------------
__global__ void __launch_bounds__(256)
vq_gather_loss(const int* __restrict__ qidx, const float* __restrict__ embed,
               const float* __restrict__ h, float* __restrict__ zout,
               float* __restrict__ partials) {
    __shared__ float red[256];
    const int b = blockIdx.x >> 6;
    const int e = blockIdx.x & 63;
    const int tid = threadIdx.x;

    float local = 0.f;
#pragma unroll
    for (int i = 0; i < T_SZ / 256; ++i) {
        const int t = i * 256 + tid;
        const int n = b * T_SZ + t;
        const int idx = qidx[n];
        const float z = embed[(size_t)idx * E_SZ + e];
        const float hv = h[(size_t)n * E_SZ + e];
        const float diff = z - hv;
        local += diff * diff;
        zout[((size_t)b * E_SZ + e) * T_SZ + t] = z;   // straight-through fwd
    }
    red[tid] = local;
    __syncthreads();
#pragma unroll
    for (int s = 128; s > 0; s >>= 1) {
        if (tid < s) red[tid] += red[tid + s];
        __syncthreads();
    }
    if (tid == 0) partials[blockIdx.x] = red[0];
}

// --------------------------------- scalar epilogues (parallel, deterministic)
__global__ void __launch_bounds__(256)
vq_final(const float* __restrict__ partials, const int* __restrict__ hist,
         float* __restrict__ out) {
    __shared__ float red[256];
    const int tid = threadIdx.x;

    // commitment loss: fixed-assignment sum of 1024 block partials
    float s = 0.f;
#pragma unroll
    for (int i = 0; i < 4; ++i) s += partials[i * 256 + tid];
    red[tid] = s;
    __syncthreads();
    for (int st = 128; st > 0; st >>= 1) {
        if (tid < st) red[tid] += red[tid + st];
        __syncthreads();
    }
    if (tid == 0) out[OUT_LOSS] = COMMIT * red[0] / (float)(N_TOK * E_SZ);
    __syncthreads();

    // log-perplexity of code usage
    const float inv_n = 1.f / (float)N_TOK;
    float ps = 0.f;
#pragma unroll
    for (int i = 0; i < 8; ++i) {
        const float p = (float)hist[i * 256 + tid] * inv_n;
        ps -= p * logf(p + 1e-10f);
    }
    red[tid] = ps;
    __syncthreads();
    for (int st = 128; st > 0; st >>= 1) {
        if (tid < st) red[tid] += red[tid + st];
        __syncthreads();
    }
    if (tid == 0) out[OUT_PPL] = red[0];

    // kldiv_r = log(K) * (N/B) * ones(B,1)
    if (tid < B_SZ) out[OUT_KL + tid] = logf((float)K_SZ) * (float)(N_TOK / B_SZ);
}

extern "C" void kernel_launch(void* const* d_in, const int* in_sizes, int n_in,
                              void* d_out, int out_size, void* d_ws, size_t ws_size,
                              hipStream_t stream) {
    const float* x      = (const float*)d_in[0];
    const float* proj_w = (const float*)d_in[1];
    const float* proj_b = (const float*)d_in[2];
    const float* embed  = (const float*)d_in[3];
    float* out = (float*)d_out;

    float* ws      = (float*)d_ws;
    float* h       = ws + WS_H;
    float* enorm   = ws + WS_ENORM;
    int*   qidx    = (int*)(ws + WS_QIDX);
    int*   hist    = (int*)(ws + WS_HIST);
    float* partial = ws + WS_PART;

    vq_zero_hist<<<(K_SZ + 255) / 256, 256, 0, stream>>>(hist);
    vq_enorm<<<(K_SZ + 255) / 256, 256, 0, stream>>>(embed, enorm);
    vq_proj_wmma<<<N_TOK / 128, 256, 0, stream>>>(x, proj_w, proj_b, h);
    vq_dist_argmin<<<N_TOK / 128, 256, 0, stream>>>(h, embed, enorm, qidx);
    vq_hist<<<N_TOK / 256, 256, 0, stream>>>(qidx, hist);
    vq_gather_loss<<<B_SZ * E_SZ, 256, 0, stream>>>(qidx, embed, h, out + OUT_Z, partial);
    vq_final<<<1, 256, 0, stream>>>(partial, hist, out);
}